// GraphSAGE_40132174414178
// MI455X (gfx1250) — compile-verified
//
#include <hip/hip_runtime.h>
#include <hip/hip_bf16.h>

typedef __attribute__((ext_vector_type(2))) float v2f;
typedef __attribute__((ext_vector_type(8))) float v8f;

#define DIM 128
#define KTOT 256
#define KCHUNK 64

// ---------------- utility kernels ----------------

__global__ __launch_bounds__(256) void zerof_kernel(float* __restrict__ p, int n) {
    int t = blockIdx.x * 256 + threadIdx.x;
    if (t < n) p[t] = 0.0f;
}

__global__ __launch_bounds__(256) void deg_kernel(const int* __restrict__ dst,
                                                  float* __restrict__ deg, int e) {
    int t = blockIdx.x * 256 + threadIdx.x;
    if (t < e) atomicAdd(&deg[dst[t]], 1.0f);
}

__global__ __launch_bounds__(256) void recip_kernel(float* __restrict__ deg, int n) {
    int t = blockIdx.x * 256 + threadIdx.x;
    if (t < n) {
        float d = deg[t];
        deg[t] = 1.0f / fmaxf(d, 1.0f);
    }
}

// one edge handled by 32 threads, float4 per thread: E*32 threads total
__global__ __launch_bounds__(256) void scatter_kernel(const float* __restrict__ h,
                                                      const int* __restrict__ src,
                                                      const int* __restrict__ dst,
                                                      float* __restrict__ agg, int e) {
    int t = blockIdx.x * 256 + threadIdx.x;
    if (t >= e * 32) return;
    int ei = t >> 5;
    int j  = t & 31;
    int s = src[ei];
    int d = dst[ei];
    const float4 v = *(const float4*)(h + (size_t)s * DIM + j * 4);
    float* p = agg + (size_t)d * DIM + j * 4;
    atomicAdd(p + 0, v.x);
    atomicAdd(p + 1, v.y);
    atomicAdd(p + 2, v.z);
    atomicAdd(p + 3, v.w);
}

// agg[row][c] *= recip[row]   (in-place -> becomes h_neigh)
__global__ __launch_bounds__(256) void scale_kernel(float* __restrict__ agg,
                                                    const float* __restrict__ recip, int total) {
    int t = blockIdx.x * 256 + threadIdx.x;
    if (t < total) agg[t] *= recip[t >> 7];
}

// ---------------- fused SAGE GEMM (fp32 WMMA) ----------------
// out = epilogue( h @ w_self + hn @ w_neigh + bias )
// viewed as [h | hn] (N x 256)  @  [w_self ; w_neigh] (256 x 128)
// each wave: one 16-row x 128-col strip (8 accumulators of 16x16)
// weight chunks staged global -> LDS via CDNA5 async-to-LDS DMA (ASYNCcnt)
__global__ __launch_bounds__(256) void sage_gemm_kernel(
    const float* __restrict__ h,       // N x 128
    const float* __restrict__ hn,      // N x 128 (scaled agg)
    const float* __restrict__ wself,   // 128 x 128 row-major (k, n)
    const float* __restrict__ wneigh,  // 128 x 128 row-major (k, n)
    const float* __restrict__ bias,    // 128
    const float* __restrict__ res,     // N x 128 or nullptr (added after relu)
    float* __restrict__ out,           // N x 128
    int n, int relu_flag)
{
    __shared__ float wbuf[KCHUNK * DIM];  // 32 KB weight chunk
    const int tid  = threadIdx.x;
    const int wave = tid >> 5;
    const int lane = tid & 31;
    const int l15  = lane & 15;
    const int lhi  = lane >> 4;  // 0 or 1

    const int rowbase = blockIdx.x * 128 + wave * 16;
    const bool active = (rowbase < n);
    const int rb = active ? rowbase : 0;  // clamp for safe (discarded) loads

    // LDS byte offset of wbuf: low 32 bits of the generic pointer
    // (LDS aperture maps addr[31:0] to the LDS offset)
    const unsigned ldsbase = (unsigned)(uintptr_t)(&wbuf[0]);

    v8f acc[8];
#pragma unroll
    for (int ct = 0; ct < 8; ++ct)
#pragma unroll
        for (int r = 0; r < 8; ++r) acc[ct][r] = 0.0f;

    for (int kc = 0; kc < KTOT; kc += KCHUNK) {
        // --- async DMA stage of 64x128 f32 weight chunk: 2048 x 16B copies ---
        const float* wsrc = (kc < DIM) ? (wself + (size_t)kc * DIM)
                                       : (wneigh + (size_t)(kc - DIM) * DIM);
#pragma unroll
        for (int i = tid; i < (KCHUNK * DIM) / 4; i += 256) {
            unsigned lofs = ldsbase + (unsigned)i * 16u;
            const float* g = wsrc + (size_t)i * 4;
            asm volatile("global_load_async_to_lds_b128 %0, %1, off"
                         :: "v"(lofs), "v"(g)
                         : "memory");
        }
        asm volatile("s_wait_asynccnt 0x0" ::: "memory");
        __syncthreads();

        const float* asrc = (kc < DIM) ? h : hn;
        const int acol0   = (kc < DIM) ? kc : (kc - DIM);

        for (int ks = 0; ks < KCHUNK; ks += 4) {
            // A fragment 16x4 f32: v0 = K{0|2}, v1 = K{1|3} (lanes 0-15 | 16-31)
            const int arow = rb + l15;
            const int ak   = acol0 + ks + lhi * 2;
            const float2 av = *(const float2*)(asrc + (size_t)arow * DIM + ak);
            v2f a;
            a.x = av.x;
            a.y = av.y;
#pragma unroll
            for (int ct = 0; ct < 8; ++ct) {
                // B fragment 4x16 f32 from LDS: v0 = row K{0|2}, v1 = row K{1|3}
                const int bk   = ks + lhi * 2;
                const int bcol = ct * 16 + l15;
                v2f b;
                b.x = wbuf[bk * DIM + bcol];
                b.y = wbuf[(bk + 1) * DIM + bcol];
                acc[ct] = __builtin_amdgcn_wmma_f32_16x16x4_f32(
                    false, a, false, b, (short)0, acc[ct], false, false);
            }
        }
        __syncthreads();
    }

    if (!active) return;

    // epilogue: +bias, relu, +residual; C layout: VGPR r -> M = r + 8*lhi, N = l15
#pragma unroll
    for (int ct = 0; ct < 8; ++ct) {
        const int col = ct * 16 + l15;
        const float bv = bias[col];
#pragma unroll
        for (int r = 0; r < 8; ++r) {
            const int row = rowbase + r + lhi * 8;
            float v = acc[ct][r] + bv;
            if (relu_flag) v = fmaxf(v, 0.0f);
            if (res) v += res[(size_t)row * DIM + col];
            out[(size_t)row * DIM + col] = v;
        }
    }
}

// ---------------- batch norm ----------------

__global__ __launch_bounds__(256) void bn_stats_kernel(const float* __restrict__ h,
                                                       float* __restrict__ cs,
                                                       float* __restrict__ csq, int n) {
    const int col = threadIdx.x & 127;
    const int rq  = threadIdx.x >> 7;  // 0 or 1
    float s = 0.0f, sq = 0.0f;
    for (int r = blockIdx.x * 2 + rq; r < n; r += gridDim.x * 2) {
        float v = h[(size_t)r * DIM + col];
        s += v;
        sq += v * v;
    }
    atomicAdd(&cs[col], s);
    atomicAdd(&csq[col], sq);
}

__global__ __launch_bounds__(128) void bn_final_kernel(const float* __restrict__ cs,
                                                       const float* __restrict__ csq,
                                                       const float* __restrict__ g,
                                                       const float* __restrict__ b,
                                                       float* __restrict__ scale,
                                                       float* __restrict__ shift, int n) {
    int c = threadIdx.x;
    float inv_n = 1.0f / (float)n;
    float mu  = cs[c] * inv_n;
    float var = csq[c] * inv_n - mu * mu;
    float sc  = g[c] * rsqrtf(var + 1e-5f);
    scale[c] = sc;
    shift[c] = b[c] - mu * sc;
}

__global__ __launch_bounds__(256) void bn_apply_kernel(float* __restrict__ h,
                                                       const float* __restrict__ scale,
                                                       const float* __restrict__ shift,
                                                       float* __restrict__ copy_out, int total) {
    int t = blockIdx.x * 256 + threadIdx.x;
    if (t >= total) return;
    int c = t & 127;
    float v = h[t] * scale[c] + shift[c];
    h[t] = v;
    if (copy_out) copy_out[t] = v;
}

// ---------------- layer norm (one wave32 per row) ----------------

__global__ __launch_bounds__(256) void ln_kernel(const float* __restrict__ h,
                                                 const float* __restrict__ g,
                                                 const float* __restrict__ b,
                                                 float* __restrict__ out, int n) {
    int wave = (blockIdx.x * 256 + threadIdx.x) >> 5;
    int lane = threadIdx.x & 31;
    if (wave >= n) return;
    const float* row = h + (size_t)wave * DIM;
    float4 v = *(const float4*)(row + lane * 4);
    float s = v.x + v.y + v.z + v.w;
#pragma unroll
    for (int off = 16; off > 0; off >>= 1) s += __shfl_xor(s, off, 32);
    float mu = s * (1.0f / 128.0f);
    float dx = v.x - mu, dy = v.y - mu, dz = v.z - mu, dw = v.w - mu;
    float sq = dx * dx + dy * dy + dz * dz + dw * dw;
#pragma unroll
    for (int off = 16; off > 0; off >>= 1) sq += __shfl_xor(sq, off, 32);
    float rs = rsqrtf(sq * (1.0f / 128.0f) + 1e-5f);
    float4 gv = *(const float4*)(g + lane * 4);
    float4 bv = *(const float4*)(b + lane * 4);
    float4 o;
    o.x = dx * rs * gv.x + bv.x;
    o.y = dy * rs * gv.y + bv.y;
    o.z = dz * rs * gv.z + bv.z;
    o.w = dw * rs * gv.w + bv.w;
    *(float4*)(out + (size_t)wave * DIM + lane * 4) = o;
}

// ---------------- host orchestration ----------------

static inline int cdiv(int a, int b) { return (a + b - 1) / b; }

extern "C" void kernel_launch(void* const* d_in, const int* in_sizes, int n_in,
                              void* d_out, int out_size, void* d_ws, size_t ws_size,
                              hipStream_t stream) {
    const float* x   = (const float*)d_in[0];
    const int* src   = (const int*)d_in[1];
    const int* dst   = (const int*)d_in[2];
    const float* w1s = (const float*)d_in[3];
    const float* w1n = (const float*)d_in[4];
    const float* b1  = (const float*)d_in[5];
    const float* w2s = (const float*)d_in[6];
    const float* w2n = (const float*)d_in[7];
    const float* b2  = (const float*)d_in[8];
    const float* w3s = (const float*)d_in[9];
    const float* w3n = (const float*)d_in[10];
    const float* b3  = (const float*)d_in[11];
    const float* bn1g = (const float*)d_in[12];
    const float* bn1b = (const float*)d_in[13];
    const float* bn2g = (const float*)d_in[14];
    const float* bn2b = (const float*)d_in[15];
    const float* lng  = (const float*)d_in[16];
    const float* lnb  = (const float*)d_in[17];

    const int N = in_sizes[0] / DIM;
    const int E = in_sizes[1];
    const int ND = N * DIM;

    float* ws = (float*)d_ws;
    float* hA    = ws;
    float* hB    = ws + (size_t)ND;
    float* agg   = ws + (size_t)2 * ND;
    float* hres  = ws + (size_t)3 * ND;
    float* recip = ws + (size_t)4 * ND;
    float* cs    = recip + N;
    float* csq   = cs + DIM;
    float* csc   = csq + DIM;
    float* csh   = csc + DIM;

    float* outp = (float*)d_out;

    const int ZB_ND = cdiv(ND, 256);
    const int SCAT  = cdiv(E * 32, 256);
    const int GEMMB = cdiv(N, 128);

    // degree -> reciprocal (computed fresh every call; deterministic)
    zerof_kernel<<<cdiv(N, 256), 256, 0, stream>>>(recip, N);
    deg_kernel<<<cdiv(E, 256), 256, 0, stream>>>(dst, recip, E);
    recip_kernel<<<cdiv(N, 256), 256, 0, stream>>>(recip, N);

    // ---- layer 1 ----
    zerof_kernel<<<ZB_ND, 256, 0, stream>>>(agg, ND);
    scatter_kernel<<<SCAT, 256, 0, stream>>>(x, src, dst, agg, E);
    scale_kernel<<<ZB_ND, 256, 0, stream>>>(agg, recip, ND);
    sage_gemm_kernel<<<GEMMB, 256, 0, stream>>>(x, agg, w1s, w1n, b1, nullptr, hA, N, 1);
    zerof_kernel<<<1, 256, 0, stream>>>(cs, 2 * DIM);
    bn_stats_kernel<<<400, 256, 0, stream>>>(hA, cs, csq, N);
    bn_final_kernel<<<1, 128, 0, stream>>>(cs, csq, bn1g, bn1b, csc, csh, N);
    bn_apply_kernel<<<ZB_ND, 256, 0, stream>>>(hA, csc, csh, hres, ND);  // hres = h after bn1

    // ---- layer 2 ----
    zerof_kernel<<<ZB_ND, 256, 0, stream>>>(agg, ND);
    scatter_kernel<<<SCAT, 256, 0, stream>>>(hA, src, dst, agg, E);
    scale_kernel<<<ZB_ND, 256, 0, stream>>>(agg, recip, ND);
    sage_gemm_kernel<<<GEMMB, 256, 0, stream>>>(hA, agg, w2s, w2n, b2, hres, hB, N, 1);
    zerof_kernel<<<1, 256, 0, stream>>>(cs, 2 * DIM);
    bn_stats_kernel<<<400, 256, 0, stream>>>(hB, cs, csq, N);
    bn_final_kernel<<<1, 128, 0, stream>>>(cs, csq, bn2g, bn2b, csc, csh, N);
    bn_apply_kernel<<<ZB_ND, 256, 0, stream>>>(hB, csc, csh, nullptr, ND);

    // ---- layer 3 ----
    zerof_kernel<<<ZB_ND, 256, 0, stream>>>(agg, ND);
    scatter_kernel<<<SCAT, 256, 0, stream>>>(hB, src, dst, agg, E);
    scale_kernel<<<ZB_ND, 256, 0, stream>>>(agg, recip, ND);
    sage_gemm_kernel<<<GEMMB, 256, 0, stream>>>(hB, agg, w3s, w3n, b3, nullptr, hA, N, 0);

    // ---- layer norm -> output ----
    ln_kernel<<<cdiv(N * 32, 256), 256, 0, stream>>>(hA, lng, lnb, outp, N);
}